// Encoding2_65128884076666
// MI455X (gfx1250) — compile-verified
//
#include <hip/hip_runtime.h>
#include <hip/hip_bf16.h>

typedef __attribute__((ext_vector_type(2))) float v2f;
typedef __attribute__((ext_vector_type(8))) float v8f;

#define BDIM 16
#define TDIM 32
#define NDIM 32768   // C*H*W = 2*128*128
#define DDIM 4096
#define NLEV 256

// ---------------------------------------------------------------------------
// Kernel 1: per-(b,t) max over N=32768 elements. 512 blocks x 256 threads.
// ---------------------------------------------------------------------------
__global__ __launch_bounds__(256) void hd_max_kernel(const float* __restrict__ data,
                                                     float* __restrict__ mout) {
  const int bt = blockIdx.x;                       // 0..511  (b*32 + t)
  const float* p = data + (size_t)bt * NDIM;
  float v = 0.0f;
  for (int i = threadIdx.x; i < NDIM; i += 256) v = fmaxf(v, p[i]);
  // wave32 reduction
  #pragma unroll
  for (int off = 16; off > 0; off >>= 1) v = fmaxf(v, __shfl_xor(v, off, 32));
  __shared__ float red[8];
  const int wave = threadIdx.x >> 5;
  if ((threadIdx.x & 31) == 0) red[wave] = v;
  __syncthreads();
  if (threadIdx.x == 0) {
    float r = red[0];
    #pragma unroll
    for (int w = 1; w < 8; ++w) r = fmaxf(r, red[w]);
    mout[bt] = r;
  }
}

// ---------------------------------------------------------------------------
// Kernel 2: per-(b,t) 256-bin histogram of round(x/m*255) in LDS (ds_add),
// written out as float counts[512][256].
// ---------------------------------------------------------------------------
__global__ __launch_bounds__(256) void hd_hist_kernel(const float* __restrict__ data,
                                                      const float* __restrict__ m,
                                                      float* __restrict__ counts) {
  const int bt = blockIdx.x;
  __shared__ unsigned bins[NLEV];
  bins[threadIdx.x] = 0u;
  __syncthreads();
  const float mm = m[bt];
  const float scale = (mm > 0.0f) ? (255.0f / mm) : 0.0f;
  const float* p = data + (size_t)bt * NDIM;
  for (int i = threadIdx.x; i < NDIM; i += 256) {
    int idx = __float2int_rn(p[i] * scale);       // RNE matches jnp.round
    idx = idx < 0 ? 0 : (idx > 255 ? 255 : idx);
    atomicAdd(&bins[idx], 1u);
  }
  __syncthreads();
  counts[(size_t)bt * NLEV + threadIdx.x] = (float)bins[threadIdx.x];
}

// ---------------------------------------------------------------------------
// Kernel 3: WMMA f32 GEMM counts[32,256] x spatial[256,16-slice] per (b, wave),
// fused temporal bind, t-reduction and sign().
//   grid = (DDIM/64, BDIM), block = 128 threads = 4 waves (wave32).
//   Each wave owns a 16-wide d slice; M dim = 16 time rows (two chunks).
//
// V_WMMA_F32_16X16X4_F32 operand layout (wave32):
//   A (16x4):  lanes 0-15 -> M=lane  (v0:K=0, v1:K=1); lanes 16-31 (v0:K=2, v1:K=3)
//   B (4x16):  lanes 0-15 -> N=lane  (v0:K=0, v1:K=1); lanes 16-31 (v0:K=2, v1:K=3)
//   C/D: vr lanes 0-15 -> M=r, N=lane; lanes 16-31 -> M=r+8, N=lane-16
// ---------------------------------------------------------------------------
__global__ __launch_bounds__(128) void hd_wmma_kernel(const float* __restrict__ counts,
                                                      const float* __restrict__ spatial,
                                                      const float* __restrict__ temporal,
                                                      float* __restrict__ out) {
  const int b     = blockIdx.y;
  const int wave  = threadIdx.x >> 5;
  const int lane  = threadIdx.x & 31;
  const int half  = lane >> 4;                     // 0 or 1 (K sub-pair)
  const int l16   = lane & 15;
  const int dbase = blockIdx.x * 64 + wave * 16;   // 16-wide d slice per wave
  const int d     = dbase + l16;

  const float* crow = counts + (size_t)b * TDIM * NLEV;   // [t][k] row-major
  float acc = 0.0f;

  #pragma unroll
  for (int tc = 0; tc < 2; ++tc) {                 // time rows 0-15, 16-31
    v8f c = {};
    const int trow = tc * 16 + l16;                // A-matrix row (M = lane%16)
    const float* a_ptr = crow + (size_t)trow * NLEV + half * 2;
    const float* b_ptr = spatial + (size_t)(half * 2) * DDIM + d;

    for (int k0 = 0; k0 < NLEV; k0 += 4) {
      // A: two contiguous counts -> one b64 load
      v2f a = *(const v2f*)(a_ptr + k0);
      // B: spatial rows k0+half*2 and k0+half*2+1, column d (stride DDIM)
      v2f bb;
      bb.x = b_ptr[(size_t)k0 * DDIM];
      bb.y = b_ptr[(size_t)(k0 + 1) * DDIM];
      c = __builtin_amdgcn_wmma_f32_16x16x4_f32(
          /*neg_a=*/false, a, /*neg_b=*/false, bb,
          /*c_mod=*/(short)0, c, /*reuse_a=*/false, /*reuse_b=*/false);
    }

    // Epilogue: this lane holds M = tc*16 + half*8 + r, column N = l16.
    #pragma unroll
    for (int r = 0; r < 8; ++r) {
      const int t = tc * 16 + half * 8 + r;
      acc += c[r] * temporal[(size_t)t * DDIM + d];
    }
  }

  // Combine the two M-halves living in lanes l16 and l16+16.
  acc += __shfl_xor(acc, 16, 32);

  if (half == 0) {
    float s = (acc > 0.0f) ? 1.0f : ((acc < 0.0f) ? -1.0f : 0.0f);
    out[(size_t)b * DDIM + d] = s;
  }
}

// ---------------------------------------------------------------------------
// Host launcher
// ---------------------------------------------------------------------------
extern "C" void kernel_launch(void* const* d_in, const int* in_sizes, int n_in,
                              void* d_out, int out_size, void* d_ws, size_t ws_size,
                              hipStream_t stream) {
  const float* data     = (const float*)d_in[0];   // [16,32,2,128,128] f32
  const float* spatial  = (const float*)d_in[1];   // [16384,4096] f32 (rows 0..255 used)
  const float* temporal = (const float*)d_in[2];   // [32,4096] f32
  float* out = (float*)d_out;                      // [16,4096] f32

  float* counts = (float*)d_ws;                    // 512*256 f32 = 512 KB
  float* maxima = counts + (size_t)BDIM * TDIM * NLEV;  // 512 f32

  hd_max_kernel <<<dim3(BDIM * TDIM), dim3(256), 0, stream>>>(data, maxima);
  hd_hist_kernel<<<dim3(BDIM * TDIM), dim3(256), 0, stream>>>(data, maxima, counts);
  hd_wmma_kernel<<<dim3(DDIM / 64, BDIM), dim3(128), 0, stream>>>(counts, spatial,
                                                                  temporal, out);
}